// LearnedSimModel_28716151341396
// MI455X (gfx1250) — compile-verified
//
#include <hip/hip_runtime.h>
#include <math.h>

#define HD 128
#define LDSTR 132          // LDS row stride (floats), padded to avoid bank conflicts
#define NNODES 20000
#define NEDGES 640000
#define NLAYERS 10
#define LNEPS 1e-5f

typedef __attribute__((ext_vector_type(2))) float v2f;
typedef __attribute__((ext_vector_type(8))) float v8f;

static __device__ __forceinline__ v8f wmma4(v2f a, v2f b, v8f c) {
  // D = A(16x4) * B(4x16) + C, fp32 exact
  return __builtin_amdgcn_wmma_f32_16x16x4_f32(false, a, false, b, (short)0, c,
                                               false, false);
}

// 32x16 output tile (two 16x16 row tiles), one K=128 chunk, f32 WMMA.
// SA: LDS activations [32][LDSTR].  W: global weights [K][HD] row-major,
// this wave uses columns cb..cb+15.
static __device__ __forceinline__ void gemm_k128(const float* __restrict__ SA,
                                                 const float* __restrict__ W,
                                                 int cb, int lane,
                                                 v8f& acc0, v8f& acc1) {
  const int n    = lane & 15;            // A row m == B col n == lane&15
  const int koff = (lane >> 4) << 1;     // 0 (lanes 0-15) or 2 (lanes 16-31)
  const float* a0 = SA + n * LDSTR + koff;
  const float* a1 = a0 + 16 * LDSTR;
  const float* wc = W + cb + n;
#pragma unroll 8
  for (int kb = 0; kb < 128; kb += 4) {
    v2f b;
    b.x = wc[(kb + koff) * HD];
    b.y = wc[(kb + koff + 1) * HD];
    v2f aA;
    aA.x = a0[kb];
    aA.y = a0[kb + 1];
    acc0 = wmma4(aA, b, acc0);
    v2f aB;
    aB.x = a1[kb];
    aB.y = a1[kb + 1];
    acc1 = wmma4(aB, b, acc1);
  }
}

// ---------------------------------------------------------------------------
// Fused per-layer edge kernel: 32 edges per block, 8 waves x 16 output cols.
//   t     = relu([xi|xj|e] @ W1 + B1)
//   e_new = e + t @ W2 + B2
//   u     = relu([xi|e_new] @ Wn1 + Bn1)
//   msg   = xi + u @ Wn2 + Bn2   -> atomic-add into hagg[dst]
//   e    <- LayerNorm(e + e_new) * g + b        (in place)
// NEDGES % 32 == 0, so no tail masking (EXEC stays all-ones for WMMA).
// ---------------------------------------------------------------------------
__global__ __launch_bounds__(256) void gnn_layer_edge(
    const float* __restrict__ h, float* __restrict__ e,
    float* __restrict__ hagg, const int* __restrict__ eidx,
    const float* __restrict__ W1, const float* __restrict__ B1,
    const float* __restrict__ W2, const float* __restrict__ B2,
    const float* __restrict__ Wn1, const float* __restrict__ Bn1,
    const float* __restrict__ Wn2, const float* __restrict__ Bn2,
    const float* __restrict__ elng, const float* __restrict__ elnb) {
  __shared__ float s_xi[32 * LDSTR];
  __shared__ float s_xj[32 * LDSTR];
  __shared__ float s_e[32 * LDSTR];
  __shared__ float s_t[32 * LDSTR];
  __shared__ float s_en[32 * LDSTR];
  __shared__ int s_dst[32];
  __shared__ float s_mu[32];
  __shared__ float s_rs[32];

  const int tid  = threadIdx.x;
  const int lane = tid & 31;
  const int wv   = tid >> 5;
  const int cb   = wv * 16;
  const int n    = lane & 15;
  const int mb   = (lane >> 4) << 3;  // 0 or 8 (C/D row base within tile)
  const int eb   = blockIdx.x * 32;

  // Gather xi = h[dst], xj = h[src], and copy the e rows into LDS (float4).
  for (int s = tid; s < 32 * 32; s += 256) {
    const int r  = s >> 5;
    const int q  = (s & 31) << 2;
    const int eg = eb + r;
    const int sj = eidx[eg];           // src -> x_j
    const int di = eidx[NEDGES + eg];  // dst -> x_i
    *(float4*)&s_xi[r * LDSTR + q] = *(const float4*)&h[(size_t)di * HD + q];
    *(float4*)&s_xj[r * LDSTR + q] = *(const float4*)&h[(size_t)sj * HD + q];
    *(float4*)&s_e[r * LDSTR + q]  = *(const float4*)&e[(size_t)eg * HD + q];
  }
  if (tid < 32) s_dst[tid] = eidx[NEDGES + eb + tid];
  __syncthreads();

  const v8f vz = {0.f, 0.f, 0.f, 0.f, 0.f, 0.f, 0.f, 0.f};

  // ---- GEMM1: t = relu([xi|xj|e] @ W1 + B1), K = 384 in three 128-chunks.
  v8f a0 = vz, a1 = vz;
  gemm_k128(s_xi, W1, cb, lane, a0, a1);
  gemm_k128(s_xj, W1 + 128 * HD, cb, lane, a0, a1);
  gemm_k128(s_e, W1 + 256 * HD, cb, lane, a0, a1);
  {
    const float bv = B1[cb + n];
#pragma unroll
    for (int v = 0; v < 8; ++v) {
      s_t[(mb + v) * LDSTR + cb + n]      = fmaxf(a0[v] + bv, 0.f);
      s_t[(16 + mb + v) * LDSTR + cb + n] = fmaxf(a1[v] + bv, 0.f);
    }
  }
  __syncthreads();

  // ---- GEMM2: e_new = e + t @ W2 + B2
  a0 = vz;
  a1 = vz;
  gemm_k128(s_t, W2, cb, lane, a0, a1);
  {
    const float bv = B2[cb + n];
#pragma unroll
    for (int v = 0; v < 8; ++v) {
      const int m0 = mb + v, m1 = 16 + mb + v;
      s_en[m0 * LDSTR + cb + n] = s_e[m0 * LDSTR + cb + n] + a0[v] + bv;
      s_en[m1 * LDSTR + cb + n] = s_e[m1 * LDSTR + cb + n] + a1[v] + bv;
    }
  }
  __syncthreads();

  // ---- GEMM3: u = relu([xi|e_new] @ Wn1 + Bn1) -> reuse s_t
  a0 = vz;
  a1 = vz;
  gemm_k128(s_xi, Wn1, cb, lane, a0, a1);
  gemm_k128(s_en, Wn1 + 128 * HD, cb, lane, a0, a1);
  {
    const float bv = Bn1[cb + n];
#pragma unroll
    for (int v = 0; v < 8; ++v) {
      s_t[(mb + v) * LDSTR + cb + n]      = fmaxf(a0[v] + bv, 0.f);
      s_t[(16 + mb + v) * LDSTR + cb + n] = fmaxf(a1[v] + bv, 0.f);
    }
  }
  __syncthreads();

  // ---- GEMM4: msg = xi + u @ Wn2 + Bn2; scatter-add into hagg[dst]
  a0 = vz;
  a1 = vz;
  gemm_k128(s_t, Wn2, cb, lane, a0, a1);
  {
    const float bv = Bn2[cb + n];
#pragma unroll
    for (int v = 0; v < 8; ++v) {
      const int m0 = mb + v, m1 = 16 + mb + v;
      const float msg0 = s_xi[m0 * LDSTR + cb + n] + a0[v] + bv;
      const float msg1 = s_xi[m1 * LDSTR + cb + n] + a1[v] + bv;
      unsafeAtomicAdd(&hagg[(size_t)s_dst[m0] * HD + cb + n], msg0);
      unsafeAtomicAdd(&hagg[(size_t)s_dst[m1] * HD + cb + n], msg1);
    }
  }

  // ---- Fused edge LayerNorm: e <- LN(e + e_new) * g + b
  __syncthreads();
  if (tid < 32) {
    float sum = 0.f, sq = 0.f;
    const float* pe = &s_e[tid * LDSTR];
    const float* pn = &s_en[tid * LDSTR];
    for (int c = 0; c < HD; ++c) {
      const float x = pe[c] + pn[c];
      sum += x;
      sq += x * x;
    }
    const float mu = sum * (1.f / HD);
    s_mu[tid] = mu;
    s_rs[tid] = rsqrtf(sq * (1.f / HD) - mu * mu + LNEPS);
  }
  __syncthreads();
  {
    const int r  = tid >> 3;
    const int c0 = (tid & 7) << 4;
    const float mu = s_mu[r], rs = s_rs[r];
    float* po = &e[(size_t)(eb + r) * HD];
#pragma unroll
    for (int j = 0; j < 16; ++j) {
      const int c   = c0 + j;
      const float x = s_e[r * LDSTR + c] + s_en[r * LDSTR + c];
      po[c] = (x - mu) * rs * elng[c] + elnb[c];
    }
  }
}

// h <- LN(h + hagg) * g + b; also re-zero hagg for the next layer.
__global__ __launch_bounds__(256) void node_update(float* __restrict__ h,
                                                   float* __restrict__ hagg,
                                                   const float* __restrict__ g,
                                                   const float* __restrict__ b) {
  const int node = blockIdx.x * 8 + (threadIdx.x >> 5);
  const int lane = threadIdx.x & 31;
  if (node >= NNODES) return;
  const size_t base = (size_t)node * HD + lane * 4;
  const float4 hv = *(const float4*)&h[base];
  const float4 av = *(const float4*)&hagg[base];
  const float v0 = hv.x + av.x, v1 = hv.y + av.y;
  const float v2 = hv.z + av.z, v3 = hv.w + av.w;
  float s = v0 + v1 + v2 + v3;
  float q = v0 * v0 + v1 * v1 + v2 * v2 + v3 * v3;
  for (int off = 16; off >= 1; off >>= 1) {
    s += __shfl_xor(s, off, 32);
    q += __shfl_xor(q, off, 32);
  }
  const float mu = s * (1.f / HD);
  const float rs = rsqrtf(q * (1.f / HD) - mu * mu + LNEPS);
  const int c = lane * 4;
  float4 o;
  o.x = (v0 - mu) * rs * g[c + 0] + b[c + 0];
  o.y = (v1 - mu) * rs * g[c + 1] + b[c + 1];
  o.z = (v2 - mu) * rs * g[c + 2] + b[c + 2];
  o.w = (v3 - mu) * rs * g[c + 3] + b[c + 3];
  *(float4*)&h[base] = o;
  const float4 z = {0.f, 0.f, 0.f, 0.f};
  *(float4*)&hagg[base] = z;
}

// Small 2-layer encoder MLP: O[r] = relu((X[r]+NZ[r])@W1+B1)@W2+B2.
template <int IND>
__global__ __launch_bounds__(128) void encoder_mlp(
    const float* __restrict__ X, const float* __restrict__ NZ,
    const float* __restrict__ W1, const float* __restrict__ B1,
    const float* __restrict__ W2, const float* __restrict__ B2,
    float* __restrict__ O, int rows) {
  __shared__ float s_in[IND];
  __shared__ float s_hid[HD];
  const int t = threadIdx.x;
  for (int rep = 0; rep < 4; ++rep) {
    const int r = blockIdx.x * 4 + rep;
    if (r >= rows) return;  // rows % 4 == 0: uniform across the block
    if (t < IND) s_in[t] = X[(size_t)r * IND + t] + NZ[(size_t)r * IND + t];
    __syncthreads();
    float a = B1[t];
#pragma unroll
    for (int k = 0; k < IND; ++k) a += s_in[k] * W1[k * HD + t];
    s_hid[t] = fmaxf(a, 0.f);
    __syncthreads();
    float o = B2[t];
#pragma unroll 4
    for (int k = 0; k < HD; ++k) o += s_hid[k] * W2[k * HD + t];
    O[(size_t)r * HD + t] = o;
    __syncthreads();
  }
}

// out[n] = relu(h[n]@W1+B1)@W2+B2 - node_noise[n, :2]
__global__ __launch_bounds__(128) void decoder_mlp(
    const float* __restrict__ h, const float* __restrict__ W1,
    const float* __restrict__ B1, const float* __restrict__ W2,
    const float* __restrict__ B2, const float* __restrict__ noise,
    float* __restrict__ out) {
  __shared__ float s_h[HD];
  __shared__ float s_t[HD];
  const int nd = blockIdx.x;
  const int t = threadIdx.x;
  s_h[t] = h[(size_t)nd * HD + t];
  __syncthreads();
  float a = B1[t];
#pragma unroll 4
  for (int k = 0; k < HD; ++k) a += s_h[k] * W1[k * HD + t];
  s_t[t] = fmaxf(a, 0.f);
  __syncthreads();
  if (t < 2) {
    float o = B2[t];
    for (int k = 0; k < HD; ++k) o += s_t[k] * W2[k * 2 + t];
    out[(size_t)nd * 2 + t] = o - noise[(size_t)nd * 16 + t];
  }
}

__global__ void zero_f32(float* __restrict__ p, size_t nelem) {
  size_t i = (size_t)blockIdx.x * blockDim.x + threadIdx.x;
  const size_t stride = (size_t)gridDim.x * blockDim.x;
  for (; i < nelem; i += stride) p[i] = 0.f;
}

extern "C" void kernel_launch(void* const* d_in, const int* in_sizes, int n_in,
                              void* d_out, int out_size, void* d_ws,
                              size_t ws_size, hipStream_t stream) {
  (void)in_sizes; (void)n_in; (void)out_size; (void)ws_size;
  const float* x      = (const float*)d_in[0];
  const float* eattr  = (const float*)d_in[1];
  const int*   eidx   = (const int*)d_in[2];
  const float* nnoise = (const float*)d_in[3];
  const float* enoise = (const float*)d_in[4];
  const float* ne_w1  = (const float*)d_in[5];
  const float* ne_b1  = (const float*)d_in[6];
  const float* ne_w2  = (const float*)d_in[7];
  const float* ne_b2  = (const float*)d_in[8];
  const float* ee_w1  = (const float*)d_in[9];
  const float* ee_b1  = (const float*)d_in[10];
  const float* ee_w2  = (const float*)d_in[11];
  const float* ee_b2  = (const float*)d_in[12];
  const float* gw1    = (const float*)d_in[13];
  const float* gb1    = (const float*)d_in[14];
  const float* gw2    = (const float*)d_in[15];
  const float* gb2    = (const float*)d_in[16];
  const float* gnw1   = (const float*)d_in[17];
  const float* gnb1   = (const float*)d_in[18];
  const float* gnw2   = (const float*)d_in[19];
  const float* gnb2   = (const float*)d_in[20];
  const float* xlng   = (const float*)d_in[21];
  const float* xlnb   = (const float*)d_in[22];
  const float* elng   = (const float*)d_in[23];
  const float* elnb   = (const float*)d_in[24];
  const float* dw1    = (const float*)d_in[25];
  const float* db1    = (const float*)d_in[26];
  const float* dw2    = (const float*)d_in[27];
  const float* db2    = (const float*)d_in[28];
  float* out = (float*)d_out;

  float* h    = (float*)d_ws;                       // [N,H]
  float* hagg = h + (size_t)NNODES * HD;            // [N,H]
  float* e    = hagg + (size_t)NNODES * HD;         // [E,H]

  zero_f32<<<512, 256, 0, stream>>>(hagg, (size_t)NNODES * HD);
  encoder_mlp<16><<<NNODES / 4, 128, 0, stream>>>(x, nnoise, ne_w1, ne_b1,
                                                  ne_w2, ne_b2, h, NNODES);
  encoder_mlp<8><<<NEDGES / 4, 128, 0, stream>>>(eattr, enoise, ee_w1, ee_b1,
                                                 ee_w2, ee_b2, e, NEDGES);
  for (int l = 0; l < NLAYERS; ++l) {
    gnn_layer_edge<<<NEDGES / 32, 256, 0, stream>>>(
        h, e, hagg, eidx, gw1, gb1, gw2, gb2, gnw1, gnb1, gnw2, gnb2,
        elng + (size_t)l * HD, elnb + (size_t)l * HD);
    node_update<<<(NNODES + 7) / 8, 256, 0, stream>>>(
        h, hagg, xlng + (size_t)l * HD, xlnb + (size_t)l * HD);
  }
  decoder_mlp<<<NNODES, 128, 0, stream>>>(h, dw1, db1, dw2, db2, nnoise, out);
}